// SphericalHarmonics_68410239091407
// MI455X (gfx1250) — compile-verified
//
#include <hip/hip_runtime.h>
#include <stdint.h>

#define L_MAX 8
#define NCOLS 81            // (L_MAX+1)^2
#define BLOCK 128           // 4 wave32s; static LDS = 128*81*4 = 41472 B

// ---------- compile-time coefficient table (no runtime transcendentals) ----------
constexpr double c_sqrt(double x) {
  double g = x > 1.0 ? x : 1.0;
  for (int i = 0; i < 64; ++i) g = 0.5 * (g + x / g);
  return g;
}
constexpr double c_fact(int n) {
  double r = 1.0;
  for (int i = 2; i <= n; ++i) r *= (double)i;
  return r;
}
struct KTab { float k[L_MAX + 1][L_MAX + 1]; };
constexpr KTab make_ktab() {
  KTab t{};
  const double FOUR_PI = 4.0 * 3.14159265358979323846;
  for (int l = 0; l <= L_MAX; ++l)
    for (int m = 0; m <= l; ++m) {
      double k = c_sqrt((2.0 * l + 1.0) / FOUR_PI * c_fact(l - m) / c_fact(l + m));
      if (m > 0) k *= c_sqrt(2.0);   // sqrt(2) folded in for m != 0
      t.k[l][m] = (float)k;
    }
  return t;
}
constexpr KTab KT = make_ktab();
constexpr float INVD[L_MAX + 1] = {0.0f, 1.0f, 1.0f/2.0f, 1.0f/3.0f, 1.0f/4.0f,
                                   1.0f/5.0f, 1.0f/6.0f, 1.0f/7.0f, 1.0f/8.0f};

// ---------- CDNA5 async LDS -> global store path (ASYNCcnt) ----------
__device__ __forceinline__ void async_store_from_lds_b128(const float* gaddr, uint32_t lds_off) {
  asm volatile("global_store_async_from_lds_b128 %0, %1, off th:TH_STORE_NT"
               :: "v"((uint64_t)(uintptr_t)gaddr), "v"(lds_off)
               : "memory");
}
__device__ __forceinline__ void async_store_from_lds_b32(const float* gaddr, uint32_t lds_off) {
  asm volatile("global_store_async_from_lds_b32 %0, %1, off th:TH_STORE_NT"
               :: "v"((uint64_t)(uintptr_t)gaddr), "v"(lds_off)
               : "memory");
}
__device__ __forceinline__ void wait_asynccnt0() {
  asm volatile("s_wait_asynccnt 0" ::: "memory");
}

// emit one (l, m) pair of outputs into this thread's LDS row
__device__ __forceinline__ void sh_emit(float* __restrict__ row, int l, int m, float p,
                                        const float* __restrict__ cmv,
                                        const float* __restrict__ smv) {
  const int c = l * l + l;
  if (m == 0) {
    row[c] = KT.k[l][0] * p;
  } else {
    const float kp = KT.k[l][m] * p;
    row[c + m] = kp * cmv[m];   // m > 0 column
    row[c - m] = kp * smv[m];   // m < 0 column
  }
}

__global__ __launch_bounds__(BLOCK)
void sh_l8_kernel(const float* __restrict__ xyz, float* __restrict__ out, int n) {
  __shared__ __align__(16) float smem[BLOCK * NCOLS];
  const int t    = (int)threadIdx.x;
  const int base = (int)blockIdx.x * BLOCK;
  const int i    = base + t;

  if (i < n) {
    const float x = xyz[3 * i + 0];
    const float y = xyz[3 * i + 1];
    const float z = xyz[3 * i + 2];

    const float rxy2 = x * x + y * y;
    const float r    = sqrtf(rxy2 + z * z);
    const float rxy  = sqrtf(rxy2);
    const float ct   = z / r;
    const float st   = rxy / r;
    const bool  ok   = rxy > 0.0f;
    const float cp   = ok ? x / rxy : 0.0f;
    const float sp   = ok ? y / rxy : 0.0f;

    // azimuthal recurrences cos(m*phi), sin(m*phi)
    float cmv[L_MAX + 1], smv[L_MAX + 1];
    cmv[0] = 1.0f; smv[0] = 0.0f;
#pragma unroll
    for (int m = 1; m <= L_MAX; ++m) {
      cmv[m] = cmv[m - 1] * cp - smv[m - 1] * sp;
      smv[m] = smv[m - 1] * cp + cmv[m - 1] * sp;
    }

    float* row = &smem[t * NCOLS];   // odd stride (81) -> conflict-free DS writes

    // associated Legendre recurrences, m-major to keep only 3 live P values
    float pmm = 1.0f;                // P[0][0]
#pragma unroll
    for (int m = 0; m <= L_MAX; ++m) {
      sh_emit(row, m, m, pmm, cmv, smv);                 // P[m][m]
      if (m < L_MAX) {
        float plm1 = pmm;                                // P[m][m]
        float pl   = (float)(2 * m + 1) * ct * pmm;      // P[m+1][m]
        sh_emit(row, m + 1, m, pl, cmv, smv);
#pragma unroll
        for (int l = m + 2; l <= L_MAX; ++l) {
          const float pn = ((float)(2 * l - 1) * ct * pl
                            - (float)(l + m - 1) * plm1) * INVD[l - m];
          plm1 = pl; pl = pn;
          sh_emit(row, l, m, pl, cmv, smv);
        }
        pmm = (float)(2 * m + 1) * st * pmm;             // P[m+1][m+1]
      }
    }
  }

  __syncthreads();   // DS writes visible before async engine reads LDS

  // Coalesced async LDS -> global: 16B/lane, 512B contiguous per wave-instruction
  int valid = n - base;
  if (valid > BLOCK) valid = BLOCK;
  if (valid > 0) {
    const int nflt = valid * NCOLS;
    const int nvec = nflt >> 2;                          // float4 units
    float* gout = out + (size_t)base * NCOLS;            // block tile base: 16B aligned
    const uint32_t lbase = (uint32_t)(uintptr_t)&smem[0];
    for (int v = t; v < nvec; v += BLOCK) {
      async_store_from_lds_b128(gout + 4 * v, lbase + 16u * (uint32_t)v);
    }
    const int rem = nflt & 3;                            // tail (only if valid % 4 != 0)
    if (t < rem) {
      const int idx = (nvec << 2) + t;
      async_store_from_lds_b32(gout + idx, lbase + 4u * (uint32_t)idx);
    }
  }
  wait_asynccnt0();  // drain ASYNCcnt before wave retires
}

extern "C" void kernel_launch(void* const* d_in, const int* in_sizes, int n_in,
                              void* d_out, int out_size, void* d_ws, size_t ws_size,
                              hipStream_t stream) {
  const float* xyz = (const float*)d_in[0];
  float* out = (float*)d_out;
  const int n = in_sizes[0] / 3;
  const int blocks = (n + BLOCK - 1) / BLOCK;
  hipLaunchKernelGGL(sh_l8_kernel, dim3(blocks), dim3(BLOCK), 0, stream, xyz, out, n);
}